// PointsRenderer_13855564497223
// MI455X (gfx1250) — compile-verified
//
#include <hip/hip_runtime.h>
#include <hip/hip_bf16.h>

#define KPTS 8
#define BLK  256
#define ITERS 8

// ---------------------------------------------------------------------------
// Pipelined renderer: async-stage idx tiles into LDS (ASYNCcnt), double
// buffered, while float4 feature gathers (LOADcnt) consume the previous tile.
// Requires nPix % (BLK*ITERS) == 0 (true for 8*512*512 pixels).
// ---------------------------------------------------------------------------
__global__ void __launch_bounds__(BLK)
points_render_pipelined(const int* __restrict__ idx,
                        const float4* __restrict__ feat4,
                        float* __restrict__ images,
                        unsigned nPix)
{
    __shared__ int lds_idx[2][BLK * KPTS];   // 2 x 8 KiB

    const unsigned tid    = threadIdx.x;
    const unsigned stride = gridDim.x * BLK;
    unsigned p = blockIdx.x * BLK + tid;

    // LDS byte offset of this thread's slot in buffer 0. Generic->int
    // truncation yields the workgroup-relative LDS offset (addr[31:0]).
    const unsigned lds0 = (unsigned)(uintptr_t)&lds_idx[0][tid * KPTS];
    const unsigned bufBytes = BLK * KPTS * 4; // 8192

    // Prologue: stage tile 0 into buffer 0 (32 B per lane = 2 x b128).
    {
        unsigned long long ga = (unsigned long long)(uintptr_t)(idx + (size_t)p * KPTS);
        asm volatile(
            "global_load_async_to_lds_b128 %0, %1, off\n\t"
            "global_load_async_to_lds_b128 %0, %1, off offset:16"
            :: "v"(lds0), "v"(ga) : "memory");
    }

    for (int i = 0; i < ITERS; ++i) {
        // Stage next tile into the other buffer. On the last iteration the
        // clamped (duplicate) tile is staged but never read.
        unsigned pn = p + stride;
        if (pn >= nPix) pn = p;
        {
            unsigned long long ga =
                (unsigned long long)(uintptr_t)(idx + (size_t)pn * KPTS);
            unsigned laddr = lds0 + ((unsigned)((i + 1) & 1)) * bufBytes;
            asm volatile(
                "global_load_async_to_lds_b128 %0, %1, off\n\t"
                "global_load_async_to_lds_b128 %0, %1, off offset:16"
                :: "v"(laddr), "v"(ga) : "memory");
        }
        // Allow the 2 freshly issued async ops to stay in flight; wait only
        // for the current tile's pair to have landed in LDS.
        asm volatile("s_wait_asynccnt 0x2" ::: "memory");

        const int* cur = &lds_idx[i & 1][tid * KPTS];
        int4 ia = *(const int4*)(cur);
        int4 ib = *(const int4*)(cur + 4);

        float ax = 0.f, ay = 0.f, az = 0.f, aw = 0.f;
        int cnt = 0;
        #define GATHER(vv)                                                    \
        {   int v  = (vv);                                                    \
            cnt   += (v > 0);                                                 \
            int gi = v < 0 ? 0 : v;            /* v<0 -> zero row */          \
            float m = v < 0 ? 0.0f : 1.0f;                                    \
            float4 f = feat4[gi];                                             \
            ax = fmaf(f.x, m, ax); ay = fmaf(f.y, m, ay);                     \
            az = fmaf(f.z, m, az); aw = fmaf(f.w, m, aw); }
        GATHER(ia.x) GATHER(ia.y) GATHER(ia.z) GATHER(ia.w)
        GATHER(ib.x) GATHER(ib.y) GATHER(ib.z) GATHER(ib.w)
        #undef GATHER

        float s = rsqrtf((float)cnt);
        float4 o;
        o.x = ax * s; o.y = ay * s; o.z = az * s; o.w = aw * s;
        *(float4*)(images + (size_t)p * 4) = o;

        p += stride;
    }
}

// ---------------------------------------------------------------------------
// Plain fallback (any size): same math, idx straight from global.
// ---------------------------------------------------------------------------
__global__ void __launch_bounds__(BLK)
points_render_plain(const int* __restrict__ idx,
                    const float4* __restrict__ feat4,
                    float* __restrict__ images,
                    unsigned nPix)
{
    unsigned p = blockIdx.x * blockDim.x + threadIdx.x;
    if (p >= nPix) return;
    const int4* ip = (const int4*)(idx + (size_t)p * KPTS);
    int4 ia = ip[0];
    int4 ib = ip[1];

    float ax = 0.f, ay = 0.f, az = 0.f, aw = 0.f;
    int cnt = 0;
    #define GATHER(vv)                                                        \
    {   int v  = (vv);                                                        \
        cnt   += (v > 0);                                                     \
        int gi = v < 0 ? 0 : v;                                               \
        float m = v < 0 ? 0.0f : 1.0f;                                        \
        float4 f = feat4[gi];                                                 \
        ax = fmaf(f.x, m, ax); ay = fmaf(f.y, m, ay);                         \
        az = fmaf(f.z, m, az); aw = fmaf(f.w, m, aw); }
    GATHER(ia.x) GATHER(ia.y) GATHER(ia.z) GATHER(ia.w)
    GATHER(ib.x) GATHER(ib.y) GATHER(ib.z) GATHER(ib.w)
    #undef GATHER

    float s = rsqrtf((float)cnt);
    float4 o;
    o.x = ax * s; o.y = ay * s; o.z = az * s; o.w = aw * s;
    *(float4*)(images + (size_t)p * 4) = o;
}

// depth[t] = zbuf[0, h, w, 0] == zbuf[t * K]
__global__ void __launch_bounds__(BLK)
depth_copy(const float* __restrict__ zbuf, float* __restrict__ dep, unsigned n)
{
    unsigned t = blockIdx.x * blockDim.x + threadIdx.x;
    if (t < n) dep[t] = zbuf[(size_t)t * KPTS];
}

extern "C" void kernel_launch(void* const* d_in, const int* in_sizes, int n_in,
                              void* d_out, int out_size, void* d_ws, size_t ws_size,
                              hipStream_t stream)
{
    (void)n_in; (void)d_ws; (void)ws_size;

    const int*   idx      = (const int*)d_in[0];
    const float* zbuf     = (const float*)d_in[1];
    const float* features = (const float*)d_in[2];
    float*       out      = (float*)d_out;

    const unsigned nPix = (unsigned)(in_sizes[0] / KPTS);      // B*H*W
    const unsigned HW   = (unsigned)(out_size - 4 * (int)nPix); // H*W (depth elems)

    float* images = out;                 // nPix * 4 floats, (B,H,W,4) flat
    float* depth  = out + (size_t)nPix * 4;

    const unsigned tileElems = BLK * ITERS;
    if (nPix % tileElems == 0) {
        unsigned blocks = nPix / tileElems;
        points_render_pipelined<<<blocks, BLK, 0, stream>>>(
            idx, (const float4*)features, images, nPix);
    } else {
        unsigned blocks = (nPix + BLK - 1) / BLK;
        points_render_plain<<<blocks, BLK, 0, stream>>>(
            idx, (const float4*)features, images, nPix);
    }

    unsigned dblocks = (HW + BLK - 1) / BLK;
    depth_copy<<<dblocks, BLK, 0, stream>>>(zbuf, depth, HW);
}